// RNN_19559281066514
// MI455X (gfx1250) — compile-verified
//
#include <hip/hip_runtime.h>

// CDNA5 / gfx1250: wave32, WMMA f32 16x16x4 path.
typedef __attribute__((ext_vector_type(2))) float v2f;
typedef __attribute__((ext_vector_type(8))) float v8f;

#define B_TOT   512
#define L_SEQ   1024
#define VOC     64
#define EMB     128
#define TILE_B  16
#define SEQ_PITCH 68    // 64 + pad (keeps float4 alignment, breaks bank aliasing)
#define H_PITCH   130   // 128 + pad (row bank advance = 2 -> conflict-light b64 reads)

// ---------------------------------------------------------------------------
// Precompute folded input weights:
//   Wc[v][e'] = sum_e W_in[e][v] * W_h[e'][e]         (W_inT @ Whx)  [64 x 128]
//   bc[e']    = b_h[e'] + sum_e b_in[e] * W_h[e'][e]
// ---------------------------------------------------------------------------
__global__ __launch_bounds__(256) void rnn_precompute(
    const float* __restrict__ W_in, const float* __restrict__ b_in,
    const float* __restrict__ W_h,  const float* __restrict__ b_h,
    float* __restrict__ Wc, float* __restrict__ bc)
{
    int idx = blockIdx.x * blockDim.x + threadIdx.x;
    if (idx >= VOC * EMB) return;
    int v  = idx >> 7;       // vocab row 0..63
    int ep = idx & 127;      // hidden col 0..127
    const float* whrow = W_h + (size_t)ep * (2 * EMB);   // W_h[e'][0..127] = Whx part
    float s = 0.f;
    for (int e = 0; e < EMB; ++e) s += W_in[e * VOC + v] * whrow[e];
    Wc[v * EMB + ep] = s;
    if (v == 0) {
        float sb = b_h[ep];
        for (int e = 0; e < EMB; ++e) sb += b_in[e] * whrow[e];
        bc[ep] = sb;
    }
}

// ---------------------------------------------------------------------------
// Fused RNN scan. One WG = 16 batch rows, 8 waves; wave w owns hidden columns
// [16w, 16w+16). Per step: C = bias; C += seq_tile(16x64) x Wc_frag;
// C += h(16x128) x Whh_frag; h_new = relu(C). Weights stay in VGPRs.
// Single barrier/step via double-buffered h and seq in LDS.
// ---------------------------------------------------------------------------
__global__ __launch_bounds__(256) void rnn_scan(
    const float* __restrict__ seq,   // [512,1024,64]
    const float* __restrict__ W_h,   // [128,256] -> Whh[e][e'] = W_h[e'*256+128+e]
    const float* __restrict__ W_out, // [64,128]
    const float* __restrict__ b_out, // [64]
    const float* __restrict__ Wc,    // [64,128]
    const float* __restrict__ bc,    // [128]
    float* __restrict__ out)         // [512,64]
{
    __shared__ float sbuf[2][TILE_B * SEQ_PITCH];
    __shared__ float hbuf[2][TILE_B * H_PITCH];

    const int tid   = threadIdx.x;
    const int wave  = tid >> 5;
    const int lane  = tid & 31;
    const int lhalf = lane >> 4;      // 0: lanes 0-15, 1: lanes 16-31
    const int lcol  = lane & 15;
    const int b0    = blockIdx.x * TILE_B;
    const int colg  = wave * 16 + lcol;   // hidden column owned by this lane

    // ---- loop-invariant B fragments, kept in registers for all 1024 steps ----
    // B layout (16x16x4 f32): lane l, vgpr i holds B[kbase + i + 2*(l>>4)][l&15]
    v2f Bc[16];                                   // folded input weights, K=64
#pragma unroll
    for (int kf = 0; kf < 16; ++kf) {
        int k0 = kf * 4 + 2 * lhalf;
        Bc[kf].x = Wc[(k0 + 0) * EMB + colg];
        Bc[kf].y = Wc[(k0 + 1) * EMB + colg];
    }
    v2f Bh[32];                                   // recurrent weights, K=128
#pragma unroll
    for (int kf = 0; kf < 32; ++kf) {
        int k0 = kf * 4 + 2 * lhalf;
        Bh[kf] = *(const v2f*)(W_h + (size_t)colg * (2 * EMB) + EMB + k0);
    }
    const float bcv = bc[colg];

    // ---- init h = 0, stage seq tile 0 ----
    for (int i = tid; i < TILE_B * H_PITCH; i += 256) hbuf[0][i] = 0.f;
    {
        int trow = tid >> 4, tc = (tid & 15) * 4;
        float4 g = *(const float4*)(seq + ((size_t)(b0 + trow) * L_SEQ + 0) * VOC + tc);
        *(float4*)&sbuf[0][trow * SEQ_PITCH + tc] = g;
    }
    __syncthreads();

#pragma unroll 1
    for (int t = 0; t < L_SEQ; ++t) {
        // prefetch seq tile t+1 into the other buffer (hides HBM latency)
        if (t + 1 < L_SEQ) {
            int trow = tid >> 4, tc = (tid & 15) * 4;
            float4 g = *(const float4*)(seq + ((size_t)(b0 + trow) * L_SEQ + (t + 1)) * VOC + tc);
            *(float4*)&sbuf[(t + 1) & 1][trow * SEQ_PITCH + tc] = g;
        }

        const float* sb = sbuf[t & 1];
        const float* hb = hbuf[t & 1];

        v8f c;
#pragma unroll
        for (int i = 0; i < 8; ++i) c[i] = bcv;   // bias pre-loaded into accumulator

        // A layout (16x4 f32): lane l, vgpr i holds A[l&15][kbase + i + 2*(l>>4)]
#pragma unroll
        for (int kf = 0; kf < 16; ++kf) {         // seq_t x Wc   (K = 64)
            int k0 = kf * 4 + 2 * lhalf;
            v2f a = *(const v2f*)&sb[lcol * SEQ_PITCH + k0];
            c = __builtin_amdgcn_wmma_f32_16x16x4_f32(false, a, false, Bc[kf],
                                                      (short)0, c, false, false);
        }
#pragma unroll
        for (int kf = 0; kf < 32; ++kf) {         // h x Whh      (K = 128)
            int k0 = kf * 4 + 2 * lhalf;
            v2f a = *(const v2f*)&hb[lcol * H_PITCH + k0];
            c = __builtin_amdgcn_wmma_f32_16x16x4_f32(false, a, false, Bh[kf],
                                                      (short)0, c, false, false);
        }

        // relu + write h_new to the other h buffer (D layout: row i+8*lhalf, col lcol)
        float* hn = hbuf[(t + 1) & 1];
#pragma unroll
        for (int i = 0; i < 8; ++i) {
            hn[(i + 8 * lhalf) * H_PITCH + colg] = fmaxf(c[i], 0.f);
        }
        __syncthreads();   // single barrier per step
    }

    // ---- epilogue: out = h_final @ W_out^T + b_out  (64 cols -> waves 0..3) ----
    if (wave < 4) {
        const float* hf = hbuf[L_SEQ & 1];
        v8f c;
#pragma unroll
        for (int i = 0; i < 8; ++i) c[i] = b_out[colg];
#pragma unroll
        for (int kf = 0; kf < 32; ++kf) {
            int k0 = kf * 4 + 2 * lhalf;
            v2f b = *(const v2f*)(W_out + (size_t)colg * EMB + k0);  // W_outT[e][v]
            v2f a = *(const v2f*)&hf[lcol * H_PITCH + k0];
            c = __builtin_amdgcn_wmma_f32_16x16x4_f32(false, a, false, b,
                                                      (short)0, c, false, false);
        }
#pragma unroll
        for (int i = 0; i < 8; ++i) {
            out[(size_t)(b0 + i + 8 * lhalf) * VOC + colg] = c[i];
        }
    }
}

// ---------------------------------------------------------------------------
extern "C" void kernel_launch(void* const* d_in, const int* in_sizes, int n_in,
                              void* d_out, int out_size, void* d_ws, size_t ws_size,
                              hipStream_t stream)
{
    const float* seq   = (const float*)d_in[0];
    const float* W_in  = (const float*)d_in[1];
    const float* b_in  = (const float*)d_in[2];
    const float* W_h   = (const float*)d_in[3];
    const float* b_h   = (const float*)d_in[4];
    const float* W_out = (const float*)d_in[5];
    const float* b_out = (const float*)d_in[6];
    float* out = (float*)d_out;

    float* Wc = (float*)d_ws;            // 64*128 floats
    float* bc = Wc + VOC * EMB;          // 128 floats

    rnn_precompute<<<(VOC * EMB + 255) / 256, 256, 0, stream>>>(W_in, b_in, W_h, b_h, Wc, bc);
    rnn_scan<<<B_TOT / TILE_B, 256, 0, stream>>>(seq, W_h, W_out, b_out, Wc, bc, out);
}